// RegionProposoalNetwork_49967649521905
// MI455X (gfx1250) — compile-verified
//
#include <hip/hip_runtime.h>
#include <stdint.h>

// ---------------- problem constants ----------------
#define NIMG     8
#define TANCH    159882
#define KTOT     4507          // 1000+1000+1000+1000+507
#define KPAD     4512          // KTOT padded to multiple of 16 (= 282*16)
#define NGROUP   282           // 16-column groups per mask row
#define NWORD    141           // 32-bit words per mask row (= NGROUP/2)
#define NLVL     5
#define POST_NMS 1000
#define NMS_THV  0.7f
#define IMG_W    800.0f
#define IMG_H    800.0f
#define MIN_SZ   1e-3f
#define BBOX_CLIP 4.135166556742356f   // log(1000/16)

__constant__ int c_lvl_n[NLVL]   = {120000, 30000, 7500, 1875, 507};
__constant__ int c_lvl_off[NLVL] = {0, 120000, 150000, 157500, 159375};
__constant__ int c_lvl_k[NLVL]   = {1000, 1000, 1000, 1000, 507};
__constant__ int c_slot0[NLVL]   = {0, 1000, 2000, 3000, 4000};

typedef float v2f __attribute__((ext_vector_type(2)));
typedef float v8f __attribute__((ext_vector_type(8)));

// monotonic float -> uint key (larger float => larger key)
__device__ __forceinline__ unsigned f2key(float f) {
  unsigned u = __float_as_uint(f);
  return (u & 0x80000000u) ? ~u : (u | 0x80000000u);
}

// CDNA5 async global->LDS copy (ASYNCcnt path), per-lane 16B
__device__ __forceinline__ void async_load_b128(unsigned lds_off, const void* gaddr) {
  asm volatile("global_load_async_to_lds_b128 %0, %1, off"
               :: "v"(lds_off), "v"(gaddr) : "memory");
}
__device__ __forceinline__ void wait_asynccnt0() {
  asm volatile("s_wait_asynccnt 0" ::: "memory");
}

// ---------------- K0: zero atomic counters ----------------
__global__ void rpn_init(int* cntGt, int* cntEq) {
  int t = threadIdx.x;
  if (t < NIMG * NLVL) { cntGt[t] = 0; cntEq[t] = 0; }
}

// ---------------- K1: per-(img,level) exact top-k via 8-bit radix select ----
__global__ void rpn_topk(const float* __restrict__ obj, int* __restrict__ top_idx,
                         int* __restrict__ cntGt, int* __restrict__ cntEq) {
  const int img = blockIdx.x / NLVL;
  const int lvl = blockIdx.x % NLVL;
  const int n = c_lvl_n[lvl], off = c_lvl_off[lvl], k = c_lvl_k[lvl];
  const float* p = obj + (size_t)img * TANCH + off;

  __shared__ unsigned hist[256];
  __shared__ unsigned sh_digit, sh_want;
  unsigned prefix = 0, want = (unsigned)k;

  for (int pass = 0; pass < 4; ++pass) {
    const int shift = 24 - 8 * pass;
    for (int b = threadIdx.x; b < 256; b += blockDim.x) hist[b] = 0;
    __syncthreads();
    for (int t = threadIdx.x; t < n; t += blockDim.x) {
      unsigned key = f2key(p[t]);
      bool ok = (pass == 0) || ((key >> (shift + 8)) == (prefix >> (shift + 8)));
      if (ok) atomicAdd(&hist[(key >> shift) & 255u], 1u);
    }
    __syncthreads();
    if (threadIdx.x == 0) {
      unsigned cum = 0; int d = 255;
      for (; d > 0; --d) { unsigned c = hist[d]; if (cum + c >= want) break; cum += c; }
      sh_digit = (unsigned)d; sh_want = want - cum;
    }
    __syncthreads();
    prefix |= (sh_digit << shift);
    want = sh_want;
    __syncthreads();
  }
  // prefix == exact key of the k-th largest; want == #equals still needed
  const unsigned thr = prefix, needEq = want, cgt = (unsigned)k - needEq;
  const int cidx = img * NLVL + lvl;
  int* dst = top_idx + (size_t)img * KTOT + c_slot0[lvl];
  for (int t = threadIdx.x; t < n; t += blockDim.x) {
    unsigned key = f2key(p[t]);
    if (key > thr) {
      int pos = atomicAdd(&cntGt[cidx], 1);
      dst[pos] = off + t;
    } else if (key == thr) {
      int e = atomicAdd(&cntEq[cidx], 1);
      if ((unsigned)e < needEq) dst[cgt + (unsigned)e] = off + t;
    }
  }
}

// ---------------- K2: gather + decode + clip + sigmoid (only K per image) ---
__global__ void rpn_decode(const float* __restrict__ obj, const float* __restrict__ deltas,
                           const float* __restrict__ anchors, const int* __restrict__ top_idx,
                           float* __restrict__ boxes_c, float* __restrict__ score,
                           float* __restrict__ skey, int* __restrict__ lvlArr) {
  int g = blockIdx.x * blockDim.x + threadIdx.x;
  if (g >= NIMG * KTOT) return;
  int img = g / KTOT, slot = g - img * KTOT;
  int lvl = slot / 1000; if (lvl > 4) lvl = 4;
  int t = top_idx[g];
  float ax1 = anchors[4*t], ay1 = anchors[4*t+1], ax2 = anchors[4*t+2], ay2 = anchors[4*t+3];
  float w = ax2 - ax1, h = ay2 - ay1, cx = ax1 + 0.5f * w, cy = ay1 + 0.5f * h;
  const float* dd = deltas + ((size_t)img * TANCH + t) * 4;
  float dx = dd[0], dy = dd[1];
  float dw = fminf(dd[2], BBOX_CLIP), dh = fminf(dd[3], BBOX_CLIP);
  float pcx = dx * w + cx, pcy = dy * h + cy;
  float pw = expf(dw) * w, ph = expf(dh) * h;
  float x1 = fminf(fmaxf(pcx - 0.5f * pw, 0.f), IMG_W);
  float y1 = fminf(fmaxf(pcy - 0.5f * ph, 0.f), IMG_H);
  float x2 = fminf(fmaxf(pcx + 0.5f * pw, 0.f), IMG_W);
  float y2 = fminf(fmaxf(pcy + 0.5f * ph, 0.f), IMG_H);
  bool valid = ((x2 - x1) >= MIN_SZ) && ((y2 - y1) >= MIN_SZ);
  float logit = obj[(size_t)img * TANCH + t];
  float sc = 1.0f / (1.0f + expf(-logit));
  boxes_c[4*g] = x1; boxes_c[4*g+1] = y1; boxes_c[4*g+2] = x2; boxes_c[4*g+3] = y2;
  score[g] = sc;
  skey[g] = valid ? sc : -__builtin_inff();   // sigmoid is monotone in logit -> same order
  lvlArr[g] = lvl;
}

// ---------------- K3: per-image bitonic sort (desc by skey) + gather --------
__global__ void rpn_sort(const float* __restrict__ skey, const float* __restrict__ boxes_c,
                         const float* __restrict__ score, const int* __restrict__ lvlArr,
                         float* __restrict__ sBox, float* __restrict__ sScore,
                         int* __restrict__ sLvl, int* __restrict__ sValid) {
  const int img = blockIdx.x;
  __shared__ unsigned long long smem[8192];   // 64 KB; CDNA5 WGP has 320 KB LDS
  for (int t = threadIdx.x; t < 8192; t += blockDim.x) {
    unsigned long long v;
    if (t < KTOT) {
      unsigned inv = ~f2key(skey[(size_t)img * KTOT + t]);   // ascending by inv == descending by key
      v = ((unsigned long long)inv << 32) | (unsigned)t;     // ties -> smaller idx first (stable-like)
    } else v = ~0ull;                                        // pad sorts last
    smem[t] = v;
  }
  __syncthreads();
  for (unsigned k2 = 2; k2 <= 8192; k2 <<= 1) {
    for (unsigned j = k2 >> 1; j > 0; j >>= 1) {
      for (unsigned t = threadIdx.x; t < 8192; t += blockDim.x) {
        unsigned pt = t ^ j;
        if (pt > t) {
          unsigned long long a = smem[t], b = smem[pt];
          bool up = ((t & k2) == 0);
          if ((a > b) == up) { smem[t] = b; smem[pt] = a; }
        }
      }
      __syncthreads();
    }
  }
  for (int t = threadIdx.x; t < KPAD; t += blockDim.x) {
    unsigned idx = (unsigned)smem[t];
    int o = img * KPAD + t;
    if (t < KTOT && idx < KTOT) {
      size_t gi = (size_t)img * KTOT + idx;
      sBox[4*o] = boxes_c[4*gi];   sBox[4*o+1] = boxes_c[4*gi+1];
      sBox[4*o+2] = boxes_c[4*gi+2]; sBox[4*o+3] = boxes_c[4*gi+3];
      sScore[o] = score[gi]; sLvl[o] = lvlArr[gi];
      float kf = skey[gi];
      sValid[o] = (kf > -3.0e38f) ? 1 : 0;          // finite <=> passed min-size filter
    } else {
      sBox[4*o] = 0.f; sBox[4*o+1] = 0.f; sBox[4*o+2] = 0.f; sBox[4*o+3] = 0.f;
      sScore[o] = 0.f; sLvl[o] = -9; sValid[o] = 0;  // level -9 never matches -> no mask bits
    }
  }
}

// ---------------- K4: IoU suppression bitmask -------------------------------
// One wave per 16 sorted rows. WMMA f32 16x16x4 computes area_i+area_j for the
// 16x16 tile (A = [area_i, 1, 0, 0], B = [1; area_j; 0; 0]); VALU does the
// intersection; ballot_w32 packs 16 mask bits per row per group.
// Cross-level IoU is exactly 0 under the reference's level-offset trick, so a
// level-equality test replaces the offset boxes entirely.
__global__ __launch_bounds__(32) void rpn_ioumask(const float* __restrict__ sBox,
                                                  const int* __restrict__ sLvl,
                                                  unsigned short* __restrict__ mask) {
  const int img  = blockIdx.x / (KPAD / 16);
  const int rb   = blockIdx.x % (KPAD / 16);
  const int lane = threadIdx.x;
  __shared__ float rowBox[16][4];
  __shared__ float rowArea[16];
  __shared__ int   rowLvl[16];
  __shared__ float colBox[16][4];
  const float* base = sBox + (size_t)img * KPAD * 4;
  const int*   lvlb = sLvl + (size_t)img * KPAD;

  if (lane < 16) {
    unsigned lds = (unsigned)(uintptr_t)(&rowBox[lane][0]);
    async_load_b128(lds, (const void*)(base + (size_t)(rb * 16 + lane) * 4));
  }
  wait_asynccnt0();
  __syncthreads();
  if (lane < 16) {
    rowArea[lane] = (rowBox[lane][2] - rowBox[lane][0]) * (rowBox[lane][3] - rowBox[lane][1]);
    rowLvl[lane]  = lvlb[rb * 16 + lane];
  }
  __syncthreads();

  const bool lo = (lane < 16);
  v2f aOp;                                   // A 16x4 layout: lanes 0-15 hold K0,K1; 16-31 hold K2,K3
  aOp.x = lo ? rowArea[lane & 15] : 0.0f;    // A[m][0] = area_i
  aOp.y = lo ? 1.0f : 0.0f;                  // A[m][1] = 1

  unsigned short* mrow = mask + ((size_t)img * KPAD + (size_t)rb * 16) * NGROUP;

  for (int ct = 0; ct < NGROUP; ++ct) {
    if (lane < 16) {
      unsigned lds = (unsigned)(uintptr_t)(&colBox[lane][0]);
      async_load_b128(lds, (const void*)(base + (size_t)(ct * 16 + lane) * 4));
    }
    if (ct + 1 < NGROUP && lane < 16)
      __builtin_prefetch(base + (size_t)((ct + 1) * 16 + lane) * 4, 0, 1);  // global_prefetch_b8
    wait_asynccnt0();
    __syncthreads();

    const int cj = lane & 15;
    const float cx1 = colBox[cj][0], cy1 = colBox[cj][1];
    const float cx2 = colBox[cj][2], cy2 = colBox[cj][3];
    const int clvl = lvlb[ct * 16 + cj];
    const float carea = (cx2 - cx1) * (cy2 - cy1);

    v2f bOp;                                 // B[0][n] = 1, B[1][n] = area_j
    bOp.x = lo ? 1.0f : 0.0f;
    bOp.y = lo ? carea : 0.0f;
    v8f c = {};
    v8f sum = __builtin_amdgcn_wmma_f32_16x16x4_f32(false, aOp, false, bOp,
                                                    (short)0, c, false, false);

#pragma unroll
    for (int v = 0; v < 8; ++v) {
      const int rloc = v + (lo ? 0 : 8);     // D reg v: lanes 0-15 -> row v, 16-31 -> row v+8
      float rx1 = rowBox[rloc][0], ry1 = rowBox[rloc][1];
      float rx2 = rowBox[rloc][2], ry2 = rowBox[rloc][3];
      float iw = fmaxf(fminf(rx2, cx2) - fmaxf(rx1, cx1), 0.0f);
      float ih = fmaxf(fminf(ry2, cy2) - fmaxf(ry1, cy1), 0.0f);
      float inter = iw * ih;
      float uni = sum[v] - inter;            // area_i + area_j - inter (from WMMA)
      bool pred = (rowLvl[rloc] == clvl) && (inter > NMS_THV * uni);
      unsigned bal = __builtin_amdgcn_ballot_w32(pred);
      if (lane == 0) mrow[(size_t)v * NGROUP + ct]       = (unsigned short)(bal & 0xFFFFu);
      if (lane == 1) mrow[(size_t)(v + 8) * NGROUP + ct] = (unsigned short)(bal >> 16);
    }
    __syncthreads();
  }
}

// ---------------- K5: sequential greedy suppression (one wave per image) ----
__global__ __launch_bounds__(32) void rpn_suppress(const unsigned short* __restrict__ mask,
                                                   unsigned* __restrict__ supp) {
  const int img = blockIdx.x;
  const int lane = threadIdx.x;
  __shared__ unsigned s[NWORD];
  for (int w = lane; w < NWORD; w += 32) s[w] = 0u;
  __syncthreads();
  const unsigned short* m = mask + (size_t)img * KPAD * NGROUP;
  for (int i = 0; i < KTOT; ++i) {
    const unsigned wi = (unsigned)i >> 5, bi = (unsigned)i & 31u;
    const bool alive = ((s[wi] >> bi) & 1u) == 0u;
    if (alive) {
      const unsigned short* row = m + (size_t)i * NGROUP;
      for (int w = lane; w < NWORD; w += 32) {
        if (w < (int)wi) continue;                       // only j > i suppress
        unsigned word = ((unsigned)row[2 * w]) | (((unsigned)row[2 * w + 1]) << 16);
        if ((unsigned)w == wi) word &= (bi == 31u) ? 0u : (0xFFFFFFFFu << (bi + 1));
        if (word) s[w] |= word;
      }
    }
    __syncthreads();
  }
  for (int w = lane; w < NWORD; w += 32) supp[img * NWORD + w] = s[w];
}

// ---------------- K6: compaction to [8,1000,5] ------------------------------
__global__ void rpn_output(const float* __restrict__ sBox, const float* __restrict__ sScore,
                           const int* __restrict__ sValid, const unsigned* __restrict__ supp,
                           float* __restrict__ out) {
  const int img = blockIdx.x;
  const int tid = threadIdx.x;
  __shared__ unsigned scan[1024];
  __shared__ unsigned carry;
  float* o = out + (size_t)img * POST_NMS * 5;
  for (int t = tid; t < POST_NMS * 5; t += blockDim.x) o[t] = 0.0f;
  if (tid == 0) carry = 0u;
  __syncthreads();
  const unsigned* sp = supp + img * NWORD;
  const int nchunk = (KTOT + 1023) / 1024;
  for (int cidx = 0; cidx < nchunk; ++cidx) {
    int j = cidx * 1024 + tid;
    unsigned flag = 0u;
    if (j < KTOT) {
      bool sup = (sp[j >> 5] >> (j & 31)) & 1u;
      if (!sup && sValid[(size_t)img * KPAD + j]) flag = 1u;
    }
    scan[tid] = flag;
    __syncthreads();
    for (int st = 1; st < 1024; st <<= 1) {
      unsigned add = (tid >= st) ? scan[tid - st] : 0u;
      __syncthreads();
      scan[tid] += add;
      __syncthreads();
    }
    unsigned pos = carry + scan[tid] - flag;
    if (flag && pos < POST_NMS) {
      size_t s = (size_t)img * KPAD + j;
      o[pos * 5 + 0] = sBox[4 * s];     o[pos * 5 + 1] = sBox[4 * s + 1];
      o[pos * 5 + 2] = sBox[4 * s + 2]; o[pos * 5 + 3] = sBox[4 * s + 3];
      o[pos * 5 + 4] = sScore[s];
    }
    __syncthreads();
    if (tid == 0) carry += scan[1023];
    __syncthreads();
  }
}

// ---------------- host-side launcher ----------------------------------------
extern "C" void kernel_launch(void* const* d_in, const int* in_sizes, int n_in,
                              void* d_out, int out_size, void* d_ws, size_t ws_size,
                              hipStream_t stream) {
  const float* obj     = (const float*)d_in[0];   // [8, T]
  const float* deltas  = (const float*)d_in[1];   // [8, T, 4]
  const float* anchors = (const float*)d_in[2];   // [T, 4]
  float* out = (float*)d_out;                     // [8, 1000, 5]
  (void)in_sizes; (void)n_in; (void)out_size; (void)ws_size;

  char* ws = (char*)d_ws;
  size_t off = 0;
  auto take = [&](size_t bytes) -> void* {
    off = (off + 255) & ~(size_t)255;
    void* p = ws + off;
    off += bytes;
    return p;
  };
  int*   top_idx = (int*)  take(sizeof(int)   * NIMG * KTOT);
  float* boxes_c = (float*)take(sizeof(float) * NIMG * KTOT * 4);
  float* score   = (float*)take(sizeof(float) * NIMG * KTOT);
  float* skey    = (float*)take(sizeof(float) * NIMG * KTOT);
  int*   lvlArr  = (int*)  take(sizeof(int)   * NIMG * KTOT);
  float* sBox    = (float*)take(sizeof(float) * NIMG * KPAD * 4);
  float* sScore  = (float*)take(sizeof(float) * NIMG * KPAD);
  int*   sLvl    = (int*)  take(sizeof(int)   * NIMG * KPAD);
  int*   sValid  = (int*)  take(sizeof(int)   * NIMG * KPAD);
  unsigned short* mask = (unsigned short*)take(sizeof(unsigned short) * (size_t)NIMG * KPAD * NGROUP);
  unsigned* supp = (unsigned*)take(sizeof(unsigned) * NIMG * NWORD);
  int* cntGt = (int*)take(sizeof(int) * NIMG * NLVL);
  int* cntEq = (int*)take(sizeof(int) * NIMG * NLVL);
  // total ws use ~22.4 MB

  rpn_init<<<1, 64, 0, stream>>>(cntGt, cntEq);
  rpn_topk<<<NIMG * NLVL, 256, 0, stream>>>(obj, top_idx, cntGt, cntEq);
  rpn_decode<<<(NIMG * KTOT + 255) / 256, 256, 0, stream>>>(obj, deltas, anchors, top_idx,
                                                            boxes_c, score, skey, lvlArr);
  rpn_sort<<<NIMG, 1024, 0, stream>>>(skey, boxes_c, score, lvlArr, sBox, sScore, sLvl, sValid);
  rpn_ioumask<<<NIMG * (KPAD / 16), 32, 0, stream>>>(sBox, sLvl, mask);
  rpn_suppress<<<NIMG, 32, 0, stream>>>(mask, supp);
  rpn_output<<<NIMG, 1024, 0, stream>>>(sBox, sScore, sValid, supp, out);
}